// PatchEncoder_2929167696024
// MI455X (gfx1250) — compile-verified
//
#include <hip/hip_runtime.h>
#include <hip/hip_bf16.h>

#define DH 128   // hidden dim (fixed by reference)

typedef __attribute__((ext_vector_type(2))) float v2f;
typedef __attribute__((ext_vector_type(8))) float v8f;

// ---------------------------------------------------------------------------
// dst[m, :] = src[idx[m], :]      (32 lanes per row, float4 each)
__global__ void __launch_bounds__(256)
gather_rows_kernel(const float* __restrict__ src, const int* __restrict__ idx,
                   float* __restrict__ dst, int M) {
    int t = blockIdx.x * blockDim.x + threadIdx.x;
    int m = t >> 5;
    if (m >= M) return;
    int c = (t & 31) * 4;
    int s = idx[m];
    *(float4*)(dst + (size_t)m * DH + c) =
        *(const float4*)(src + (size_t)s * DH + c);
}

// h[m, :] += src[idx[m], :]
__global__ void __launch_bounds__(256)
add_gather_kernel(float* __restrict__ h, const float* __restrict__ src,
                  const int* __restrict__ idx, int M) {
    int t = blockIdx.x * blockDim.x + threadIdx.x;
    int m = t >> 5;
    if (m >= M) return;
    int c = (t & 31) * 4;
    int s = idx[m];
    float4 a = *(float4*)(h + (size_t)m * DH + c);
    float4 b = *(const float4*)(src + (size_t)s * DH + c);
    a.x += b.x; a.y += b.y; a.z += b.z; a.w += b.w;
    *(float4*)(h + (size_t)m * DH + c) = a;
}

// sums[idx[m], :] += x[m, :]   (fp32 atomics, resolve in L2)
__global__ void __launch_bounds__(256)
seg_sum_kernel(const float* __restrict__ x, const int* __restrict__ idx,
               float* __restrict__ sums, int M) {
    int t = blockIdx.x * blockDim.x + threadIdx.x;
    int m = t >> 5;
    if (m >= M) return;
    int c = (t & 31) * 4;
    int s = idx[m];
    float4 v = *(const float4*)(x + (size_t)m * DH + c);
    float* o = sums + (size_t)s * DH + c;
    atomicAdd(o + 0, v.x);
    atomicAdd(o + 1, v.y);
    atomicAdd(o + 2, v.z);
    atomicAdd(o + 3, v.w);
}

__global__ void __launch_bounds__(256)
seg_count_kernel(const int* __restrict__ idx, float* __restrict__ cnt, int M) {
    int m = blockIdx.x * blockDim.x + threadIdx.x;
    if (m < M) atomicAdd(&cnt[idx[m]], 1.0f);
}

// dst[s, :] = sums[s, :] / max(cnt[s], 1)   (dst may alias sums)
__global__ void __launch_bounds__(256)
mean_kernel(float* __restrict__ dst, const float* sums, const float* __restrict__ cnt, int S) {
    int t = blockIdx.x * blockDim.x + threadIdx.x;
    int s = t >> 5;
    if (s >= S) return;
    int c = (t & 31) * 4;
    float inv = 1.0f / fmaxf(cnt[s], 1.0f);
    float4 v = *(const float4*)(sums + (size_t)s * DH + c);
    v.x *= inv; v.y *= inv; v.z *= inv; v.w *= inv;
    *(float4*)(dst + (size_t)s * DH + c) = v;
}

// GINE message passing: agg[dst[k], :] += relu(h[src[k], :] + edge_attr[emap[k], :])
__global__ void __launch_bounds__(256)
edge_msg_kernel(const float* __restrict__ h, const float* __restrict__ edge_attr,
                const int* __restrict__ emap, const int* __restrict__ srcI,
                const int* __restrict__ dstI, float* __restrict__ agg, int EC) {
    long long t = (long long)blockIdx.x * blockDim.x + threadIdx.x;
    int k = (int)(t >> 5);
    if (k >= EC) return;
    int c = ((int)t & 31) * 4;
    float4 hv = *(const float4*)(h + (size_t)srcI[k] * DH + c);
    float4 ev = *(const float4*)(edge_attr + (size_t)emap[k] * DH + c);
    float m0 = fmaxf(hv.x + ev.x, 0.0f);
    float m1 = fmaxf(hv.y + ev.y, 0.0f);
    float m2 = fmaxf(hv.z + ev.z, 0.0f);
    float m3 = fmaxf(hv.w + ev.w, 0.0f);
    float* o = agg + (size_t)dstI[k] * DH + c;
    atomicAdd(o + 0, m0);
    atomicAdd(o + 1, m1);
    atomicAdd(o + 2, m2);
    atomicAdd(o + 3, m3);
}

// ---------------------------------------------------------------------------
// Fused GEMM:  out = act( ((alpha*A + Agg) @ W) + bias )
//   A, Agg, out are [M, 128]; W is [128, 128] row-major (k-major); in-place
//   (out==A) is safe: each wave writes exactly the rows it read, after the
//   K loop. Uses V_WMMA_F32_16X16X4_F32 (exact fp32 path).
//
// Each wave owns TWO 16-row tiles (32 rows): the 8 B-fragment LDS loads per
// K-step feed 16 WMMAs (2x reuse). HAS_AGG is a template parameter so the
// hot loop has no divergent branch (no exec-mask save/restore).
//
// LDS holds W in fragment order: block [kt][n] of 4x16 values stored at
//   sW[(kt*8 + n)*64 + (k&3)*16 + (c&15)]  -> B-fragment reads are
//   bank-conflict-free (half-waves touch disjoint 16-float spans).
template<bool HAS_AGG, bool RELU>
__global__ void __launch_bounds__(256)
gemm128_wmma(const float* A, const float* Agg,
             const float* __restrict__ epsvec, int eps_idx,
             const float* __restrict__ W, const float* __restrict__ bias,
             float* out, int M) {
    __shared__ float sW[DH * DH];   // 64 KB, fragment-ordered
    const int tid = threadIdx.x;

    // cooperative stage of W into LDS (float4 per thread, 16 iters)
    for (int i = tid; i < DH * (DH / 4); i += 256) {
        int r  = i >> 5;          // 0..127 (k)
        int c4 = (i & 31) * 4;    // 0..124 (n-col, 4-aligned)
        float4 w = *(const float4*)(W + (size_t)r * DH + c4);
        int dest = (((r >> 2) * 8 + (c4 >> 4)) * 64) + (r & 3) * 16 + (c4 & 15);
        *(float4*)(sW + dest) = w;
    }
    __syncthreads();

    const int wave = tid >> 5;
    const int lane = tid & 31;
    const int row0 = (blockIdx.x * 8 + wave) * 32;
    if (row0 >= M) return;              // wave-uniform: EXEC all-ones for WMMA
    const bool t1 = (row0 + 16) < M;    // wave-uniform second-tile validity

    float alpha = 1.0f;
    if (HAS_AGG) alpha = 1.0f + epsvec[eps_idx];

    // A fragment addressing (32-bit A 16x4): lane holds row = lane&15,
    // k = ak0, ak0+1 with ak0 = (lane>>4)*2
    const int ak0 = (lane >> 4) * 2;
    const int r0  = row0 + (lane & 15);
    const int r1  = t1 ? (row0 + 16 + (lane & 15)) : r0;  // clamp keeps loads in-bounds
    // B fragment addressing within a 4x16 LDS block
    const int boff = (lane >> 4) * 32 + (lane & 15);

    const float* a0p = A + (size_t)r0 * DH + ak0;
    const float* a1p = A + (size_t)r1 * DH + ak0;
    const float* g0p = HAS_AGG ? (Agg + (size_t)r0 * DH + ak0) : nullptr;
    const float* g1p = HAS_AGG ? (Agg + (size_t)r1 * DH + ak0) : nullptr;

    v8f accA[8] = {};
    v8f accB[8] = {};

    for (int kt = 0; kt < 32; ++kt) {
        const int kk = kt * 4;
        v2f a0, a1;
        a0.x = a0p[kk]; a0.y = a0p[kk + 1];
        a1.x = a1p[kk]; a1.y = a1p[kk + 1];
        if (HAS_AGG) {
            a0.x = alpha * a0.x + g0p[kk];
            a0.y = alpha * a0.y + g0p[kk + 1];
            a1.x = alpha * a1.x + g1p[kk];
            a1.y = alpha * a1.y + g1p[kk + 1];
        }
        const float* wk = &sW[kt * 512];   // 8 n-tiles * 64 floats
        #pragma unroll
        for (int n = 0; n < 8; ++n) {
            v2f b;
            b.x = wk[n * 64 + boff];
            b.y = wk[n * 64 + boff + 16];
            accA[n] = __builtin_amdgcn_wmma_f32_16x16x4_f32(
                false, a0, false, b, (short)0, accA[n], false, false);
            accB[n] = __builtin_amdgcn_wmma_f32_16x16x4_f32(
                false, a1, false, b, (short)0, accB[n], false, false);
        }
    }

    // C/D layout: VGPR v holds row v + 8*(lane>>4), col = lane&15 (per n-tile)
    const int rhi = (lane >> 4) * 8;
    #pragma unroll
    for (int n = 0; n < 8; ++n) {
        const int col = n * 16 + (lane & 15);
        const float bv = bias[col];
        #pragma unroll
        for (int v = 0; v < 8; ++v) {
            float val = accA[n][v] + bv;
            if (RELU) val = fmaxf(val, 0.0f);
            out[(size_t)(row0 + rhi + v) * DH + col] = val;
        }
        if (t1) {
            #pragma unroll
            for (int v = 0; v < 8; ++v) {
                float val = accB[n][v] + bv;
                if (RELU) val = fmaxf(val, 0.0f);
                out[(size_t)(row0 + 16 + rhi + v) * DH + col] = val;
            }
        }
    }
}

// ---------------------------------------------------------------------------
extern "C" void kernel_launch(void* const* d_in, const int* in_sizes, int n_in,
                              void* d_out, int out_size, void* d_ws, size_t ws_size,
                              hipStream_t stream) {
    (void)n_in; (void)ws_size;
    const float* x         = (const float*)d_in[0];
    const float* edge_attr = (const float*)d_in[1];
    const int*   nmap      = (const int*)d_in[2];
    const int*   emap      = (const int*)d_in[3];
    const int*   comb      = (const int*)d_in[4];
    const int*   batch     = (const int*)d_in[5];
    // d_in[6] = num_patches (device scalar) -- P derived from out_size instead
    const float* W_gnn     = (const float*)d_in[7];
    const float* b_gnn     = (const float*)d_in[8];
    const float* eps       = (const float*)d_in[9];
    const float* W_u       = (const float*)d_in[10];
    const float* b_u       = (const float*)d_in[11];

    const int N  = in_sizes[0] / DH;
    const int M  = in_sizes[2];
    const int EC = in_sizes[3];
    const int P  = out_size / DH;
    const int L  = in_sizes[9];
    const int* srcI = comb;
    const int* dstI = comb + EC;

    // workspace layout (floats): ~233 MB total
    float* h        = (float*)d_ws;                    // M*128
    float* agg      = h + (size_t)M * DH;              // M*128
    float* nodesum  = agg + (size_t)M * DH;            // N*128
    float* patchsum = nodesum + (size_t)N * DH;        // P*128
    float* proj     = patchsum + (size_t)P * DH;       // P*128
    float* patchcnt = proj + (size_t)P * DH;           // P
    float* nodecnt  = patchcnt + P;                    // N

    const int T = 256;
    auto rb = [&](long long total) { return (int)((total + T - 1) / T); };
    auto gemm_blocks = [](int rows) {
        int strips = (rows + 31) / 32;       // 32 rows per wave
        return (strips + 7) / 8;             // 8 waves per block
    };

    // segment counts (index arrays are layer-invariant -> compute once)
    hipMemsetAsync(patchcnt, 0, sizeof(float) * (size_t)(P + N), stream);
    seg_count_kernel<<<rb(M), T, 0, stream>>>(batch, patchcnt, M);
    seg_count_kernel<<<rb(M), T, 0, stream>>>(nmap, nodecnt, M);

    // h = x[nmap]
    gather_rows_kernel<<<rb((long long)M * 32), T, 0, stream>>>(x, nmap, h, M);

    for (int i = 0; i < L; ++i) {
        if (i > 0) {
            // sub = seg_mean(h, batch, P); h += relu(sub @ W_u + b_u)[batch]
            hipMemsetAsync(patchsum, 0, sizeof(float) * (size_t)P * DH, stream);
            seg_sum_kernel<<<rb((long long)M * 32), T, 0, stream>>>(h, batch, patchsum, M);
            mean_kernel<<<rb((long long)P * 32), T, 0, stream>>>(patchsum, patchsum, patchcnt, P);
            gemm128_wmma<false, true><<<gemm_blocks(P), 256, 0, stream>>>(
                patchsum, nullptr, nullptr, 0,
                W_u + (size_t)(i - 1) * DH * DH, b_u + (size_t)(i - 1) * DH,
                proj, P);
            add_gather_kernel<<<rb((long long)M * 32), T, 0, stream>>>(h, proj, batch, M);

            // h = seg_mean(h, nmap, N)[nmap]
            hipMemsetAsync(nodesum, 0, sizeof(float) * (size_t)N * DH, stream);
            seg_sum_kernel<<<rb((long long)M * 32), T, 0, stream>>>(h, nmap, nodesum, M);
            mean_kernel<<<rb((long long)N * 32), T, 0, stream>>>(nodesum, nodesum, nodecnt, N);
            gather_rows_kernel<<<rb((long long)M * 32), T, 0, stream>>>(nodesum, nmap, h, M);
        }

        // agg = segment_sum(relu(h[src] + e), dst)
        hipMemsetAsync(agg, 0, sizeof(float) * (size_t)M * DH, stream);
        edge_msg_kernel<<<rb((long long)EC * 32), T, 0, stream>>>(
            h, edge_attr, emap, srcI, dstI, agg, EC);

        // h = ((1+eps)*h + agg) @ W_gnn[i] + b_gnn[i]   (in place)
        gemm128_wmma<true, false><<<gemm_blocks(M), 256, 0, stream>>>(
            h, agg, eps, i,
            W_gnn + (size_t)i * DH * DH, b_gnn + (size_t)i * DH,
            h, M);
    }

    // out = seg_mean(h, batch, P)
    hipMemsetAsync(patchsum, 0, sizeof(float) * (size_t)P * DH, stream);
    seg_sum_kernel<<<rb((long long)M * 32), T, 0, stream>>>(h, batch, patchsum, M);
    mean_kernel<<<rb((long long)P * 32), T, 0, stream>>>((float*)d_out, patchsum, patchcnt, P);
}